// GINconv_two_aggregators_Net_67508295958858
// MI455X (gfx1250) — compile-verified
//
#include <hip/hip_runtime.h>

typedef __attribute__((ext_vector_type(2))) float v2f;
typedef __attribute__((ext_vector_type(8))) float v8f;

#define N_NODES  100000
#define N_EDGES  1600000
#define N_GRAPHS 1000
#define IN_CH    64
#define HID      32

// ---------------------------------------------------------------------------
// Zero-fill (float4 grid-stride). Needed because harness poisons ws with 0xAA.
// ---------------------------------------------------------------------------
__global__ void zero_kernel(float* __restrict__ p, size_t n4) {
  size_t i      = (size_t)blockIdx.x * blockDim.x + threadIdx.x;
  size_t stride = (size_t)gridDim.x * blockDim.x;
  float4 z = {0.0f, 0.0f, 0.0f, 0.0f};
  for (; i < n4; i += stride) reinterpret_cast<float4*>(p)[i] = z;
}

// ---------------------------------------------------------------------------
// Edge-parallel gather + atomic scatter-add: agg[dst] += x[src].
// ei layout: row 0 = src (2,E row-major), row 1 = dst.  C = 64 or 32.
// Working set is L2-resident (25.6 MB << 192 MB), so f32 atomics hit L2.
// ---------------------------------------------------------------------------
template <int C>
__global__ void scatter_add_kernel(const float* __restrict__ x,
                                   const int* __restrict__ ei,
                                   float* __restrict__ agg) {
  constexpr int G = C / 4;  // float4 groups per edge
  long long t = (long long)blockIdx.x * blockDim.x + threadIdx.x;
  int e = (int)(t / G);
  int g = (int)(t % G);
  if (e >= N_EDGES) return;
  int s = ei[e];
  int d = ei[N_EDGES + e];
  const float4 v = *reinterpret_cast<const float4*>(x + (size_t)s * C + g * 4);
  float* p = agg + (size_t)d * C + g * 4;
  atomicAdd(p + 0, v.x);
  atomicAdd(p + 1, v.y);
  atomicAdd(p + 2, v.z);
  atomicAdd(p + 3, v.w);
}

// ---------------------------------------------------------------------------
// Fused GIN MLP with fp32 WMMA (V_WMMA_F32_16X16X4_F32):
//   out = relu?( relu((x [+ agg]) @ W1 + b1) @ W2 + b2 )
// One wave32 handles a 16-node tile. 4 waves / block.
//
// VGPR layouts per CDNA5 ISA (wave32):
//   A 16x4  : lane m=l&15, kb=2*(l>>4); A[0]=h[m][k0+kb], A[1]=h[m][k0+kb+1]
//   B 4x16  : lane n=l&15, kb=2*(l>>4); B[0]=W[k0+kb][n], B[1]=W[k0+kb+1][n]
//   C/D     : reg i, lanes 0-15 -> row i, lanes 16-31 -> row i+8; col = n
// ---------------------------------------------------------------------------
template <int CIN, bool ADD_AGG>
__global__ __launch_bounds__(128) void gin_mlp_wmma_kernel(
    const float* __restrict__ x, const float* __restrict__ agg,
    const float* __restrict__ W1, const float* __restrict__ b1,
    const float* __restrict__ W2, const float* __restrict__ b2,
    float* __restrict__ out, int out_stride, int col_off, int relu_out) {
  constexpr int WAVES = 4;
  __shared__ float sW1[CIN * HID];
  __shared__ float sW2[HID * HID];
  __shared__ float sB1[HID];
  __shared__ float sB2[HID];
  __shared__ float sT1[WAVES][16 * 33];  // stride 33: conflict-free rows

  // Stage weights in LDS (all threads participate, then single block barrier).
  for (int i = threadIdx.x; i < CIN * HID; i += blockDim.x) sW1[i] = W1[i];
  for (int i = threadIdx.x; i < HID * HID; i += blockDim.x) sW2[i] = W2[i];
  if (threadIdx.x < HID) {
    sB1[threadIdx.x] = b1[threadIdx.x];
    sB2[threadIdx.x] = b2[threadIdx.x];
  }
  __syncthreads();

  const int wave    = threadIdx.x >> 5;
  const int lane    = threadIdx.x & 31;
  const int tile    = blockIdx.x * WAVES + wave;
  const int rowbase = tile * 16;
  if (rowbase >= N_NODES) return;  // past the only block-wide barrier

  const int m      = lane & 15;        // A-row / B,D-col within tile
  const int kb     = (lane >> 4) * 2;  // K sub-offset for this lane half
  const int rowoff = (lane >> 4) * 8;  // D row offset for this lane half

  const float* xrow = x + (size_t)(rowbase + m) * CIN;
  const float* arow = ADD_AGG ? (agg + (size_t)(rowbase + m) * CIN) : nullptr;

  // ---- GEMM1: t1 = relu((x+agg) @ W1 + b1), two 16-col halves -------------
  v8f acc0 = {};
  v8f acc1 = {};
#pragma unroll
  for (int k0 = 0; k0 < CIN; k0 += 4) {
    float ax = xrow[k0 + kb + 0];
    float ay = xrow[k0 + kb + 1];
    if (ADD_AGG) {
      ax += arow[k0 + kb + 0];
      ay += arow[k0 + kb + 1];
    }
    v2f A = {ax, ay};
    v2f B0 = {sW1[(k0 + kb) * HID + m], sW1[(k0 + kb + 1) * HID + m]};
    v2f B1v = {sW1[(k0 + kb) * HID + 16 + m], sW1[(k0 + kb + 1) * HID + 16 + m]};
    acc0 = __builtin_amdgcn_wmma_f32_16x16x4_f32(false, A, false, B0, (short)0,
                                                 acc0, false, false);
    acc1 = __builtin_amdgcn_wmma_f32_16x16x4_f32(false, A, false, B1v, (short)0,
                                                 acc1, false, false);
  }

  // bias + ReLU, stage t1 into LDS (intra-wave LDS ops are in-order; no
  // cross-wave sharing, so no block barrier needed).
  {
    float bv0 = sB1[m];
    float bv1 = sB1[16 + m];
    float* t = &sT1[wave][0];
#pragma unroll
    for (int i = 0; i < 8; ++i) {
      t[(i + rowoff) * 33 + m]      = fmaxf(acc0[i] + bv0, 0.0f);
      t[(i + rowoff) * 33 + 16 + m] = fmaxf(acc1[i] + bv1, 0.0f);
    }
  }
  __builtin_amdgcn_wave_barrier();  // keep compiler from reordering LDS ops

  // ---- GEMM2: out = t1 @ W2 + b2 ------------------------------------------
  v8f o0 = {};
  v8f o1 = {};
  const float* t = &sT1[wave][0];
#pragma unroll
  for (int k0 = 0; k0 < HID; k0 += 4) {
    v2f A = {t[m * 33 + k0 + kb], t[m * 33 + k0 + kb + 1]};
    v2f B0 = {sW2[(k0 + kb) * HID + m], sW2[(k0 + kb + 1) * HID + m]};
    v2f B1v = {sW2[(k0 + kb) * HID + 16 + m], sW2[(k0 + kb + 1) * HID + 16 + m]};
    o0 = __builtin_amdgcn_wmma_f32_16x16x4_f32(false, A, false, B0, (short)0,
                                               o0, false, false);
    o1 = __builtin_amdgcn_wmma_f32_16x16x4_f32(false, A, false, B1v, (short)0,
                                               o1, false, false);
  }

  const float c0 = sB2[m];
  const float c1 = sB2[16 + m];
  float* obase = out + (size_t)rowbase * out_stride + col_off;
#pragma unroll
  for (int i = 0; i < 8; ++i) {
    float r0 = o0[i] + c0;
    float r1 = o1[i] + c1;
    if (relu_out) {
      r0 = fmaxf(r0, 0.0f);
      r1 = fmaxf(r1, 0.0f);
    }
    obase[(size_t)(i + rowoff) * out_stride + m]      = r0;
    obase[(size_t)(i + rowoff) * out_stride + 16 + m] = r1;
  }
}

// ---------------------------------------------------------------------------
// Graph pooling: pooled[batch[node]][c] += h[node][c]
// ---------------------------------------------------------------------------
__global__ void pool_scatter_kernel(const float* __restrict__ h,
                                    const int* __restrict__ batch,
                                    float* __restrict__ pooled) {
  long long t = (long long)blockIdx.x * blockDim.x + threadIdx.x;
  int node = (int)(t >> 5);
  int c    = (int)(t & 31);
  if (node >= N_NODES) return;
  atomicAdd(&pooled[(size_t)batch[node] * HID + c],
            h[(size_t)node * HID + c]);
}

__global__ void final_linear_kernel(const float* __restrict__ pooled,
                                    const float* __restrict__ lw,
                                    const float* __restrict__ lb,
                                    float* __restrict__ out) {
  int g = blockIdx.x * blockDim.x + threadIdx.x;
  if (g >= N_GRAPHS) return;
  float s = lb[0];
#pragma unroll
  for (int c = 0; c < HID; ++c) s += pooled[g * HID + c] * lw[c];
  out[g] = s;
}

// ---------------------------------------------------------------------------
extern "C" void kernel_launch(void* const* d_in, const int* in_sizes, int n_in,
                              void* d_out, int out_size, void* d_ws,
                              size_t ws_size, hipStream_t stream) {
  const float* x    = (const float*)d_in[0];
  const int* eiL    = (const int*)d_in[1];
  const int* eiG    = (const int*)d_in[2];
  const int* batch  = (const int*)d_in[3];
  // param blocks: c11=4..7, c12=8..11, m1=12..15, c21=16..19, c22=20..23,
  //               m2=24..27, lin_W=28, lin_b=29
  auto P = [&](int i) { return (const float*)d_in[i]; };

  // Workspace layout (floats): B0,B1,B2 are [N,64]; B3 is [N,32]; PL [G,32].
  float* B0 = (float*)d_ws;
  float* B1 = B0 + (size_t)N_NODES * 64;
  float* B2 = B1 + (size_t)N_NODES * 64;
  float* B3 = B2 + (size_t)N_NODES * 64;
  float* PL = B3 + (size_t)N_NODES * 32;

  auto zero = [&](float* p, size_t n) {
    size_t n4 = n / 4;
    int blocks = (int)((n4 + 255) / 256);
    if (blocks > 65535) blocks = 65535;  // grid-stride covers the rest
    zero_kernel<<<blocks, 256, 0, stream>>>(p, n4);
  };

  const int tiles    = N_NODES / 16;          // 6250 (exact)
  const int ginGrid  = (tiles + 3) / 4;       // 4 waves / block
  const int sc64Grid = (int)(((long long)N_EDGES * 16 + 255) / 256);
  const int sc32Grid = (int)(((long long)N_EDGES * 8 + 255) / 256);

  // ---- Layer 1: two GINs on x (IN_CH=64) ----------------------------------
  zero(B0, (size_t)N_NODES * 64);
  zero(B1, (size_t)N_NODES * 64);
  scatter_add_kernel<64><<<sc64Grid, 256, 0, stream>>>(x, eiL, B0);
  scatter_add_kernel<64><<<sc64Grid, 256, 0, stream>>>(x, eiG, B1);
  // x1 -> cat1[:, 0:32], x2 -> cat1[:, 32:64]  (cat1 = B2)
  gin_mlp_wmma_kernel<64, true><<<ginGrid, 128, 0, stream>>>(
      x, B0, P(4), P(5), P(6), P(7), B2, 64, 0, 0);
  gin_mlp_wmma_kernel<64, true><<<ginGrid, 128, 0, stream>>>(
      x, B1, P(8), P(9), P(10), P(11), B2, 64, 32, 0);
  // h = mlp2(cat1)  (h = B3, [N,32])
  gin_mlp_wmma_kernel<64, false><<<ginGrid, 128, 0, stream>>>(
      B2, nullptr, P(12), P(13), P(14), P(15), B3, 32, 0, 0);

  // ---- Layer 2: two GINs on h (HID=32) ------------------------------------
  zero(B0, (size_t)N_NODES * 32);
  zero(B1, (size_t)N_NODES * 32);
  scatter_add_kernel<32><<<sc32Grid, 256, 0, stream>>>(B3, eiL, B0);
  scatter_add_kernel<32><<<sc32Grid, 256, 0, stream>>>(B3, eiG, B1);
  // relu(gin) -> cat2 halves (cat2 = B2)
  gin_mlp_wmma_kernel<32, true><<<ginGrid, 128, 0, stream>>>(
      B3, B0, P(16), P(17), P(18), P(19), B2, 64, 0, 1);
  gin_mlp_wmma_kernel<32, true><<<ginGrid, 128, 0, stream>>>(
      B3, B1, P(20), P(21), P(22), P(23), B2, 64, 32, 1);
  // h2 = mlp2(cat2)  (reuse B3)
  gin_mlp_wmma_kernel<64, false><<<ginGrid, 128, 0, stream>>>(
      B2, nullptr, P(24), P(25), P(26), P(27), B3, 32, 0, 0);

  // ---- Pool + final linear ------------------------------------------------
  zero(PL, (size_t)N_GRAPHS * 32);
  pool_scatter_kernel<<<(int)(((long long)N_NODES * 32 + 255) / 256), 256, 0,
                        stream>>>(B3, batch, PL);
  final_linear_kernel<<<(N_GRAPHS + 255) / 256, 256, 0, stream>>>(
      PL, P(28), P(29), (float*)d_out);
}